// PtModule_73667279061090
// MI455X (gfx1250) — compile-verified
//
#include <hip/hip_runtime.h>

// Streaming workload: x (4096x8192 fp32) -> {x+1, x-1, x*2, x/2, sum(x)}.
// ~671 MB HBM traffic => memory-bound, ~29us floor at 23.3 TB/s.
// This revision: 32-bit GVS addressing (no 64-bit VALU address math in the
// hot loop) + 8-deep b128 NT load clause per thread for MLP.

typedef float v4f __attribute__((ext_vector_type(4)));

#define BLOCK 256
#define WAVES_PER_BLOCK (BLOCK / 32)
#define UNROLL 8
#define CHUNK (BLOCK * UNROLL)   // v4f elements per block-tile (2048)

__global__ __launch_bounds__(BLOCK) void fused_stream_kernel(
    const float* __restrict__ x,
    float* __restrict__ o1, float* __restrict__ o2,
    float* __restrict__ o3, float* __restrict__ o4,
    float* __restrict__ partials,
    unsigned n4, unsigned n)
{
    const v4f* __restrict__ xv  = (const v4f*)x;
    v4f* __restrict__ o1v = (v4f*)o1;
    v4f* __restrict__ o2v = (v4f*)o2;
    v4f* __restrict__ o3v = (v4f*)o3;
    v4f* __restrict__ o4v = (v4f*)o4;

    float acc = 0.0f;

    // ---- Full tiles: 8 independent b128 NT loads, then 32 b128 NT stores ----
    const unsigned n4_full     = n4 - (n4 % CHUNK);
    const unsigned tile_stride = gridDim.x * CHUNK;

    for (unsigned base = blockIdx.x * CHUNK; base < n4_full; base += tile_stride) {
        const unsigned i0 = base + threadIdx.x;

        v4f v[UNROLL];
        #pragma unroll
        for (int u = 0; u < UNROLL; ++u)
            v[u] = __builtin_nontemporal_load(xv + (i0 + u * BLOCK));

        #pragma unroll
        for (int u = 0; u < UNROLL; ++u) {
            const unsigned i = i0 + u * BLOCK;
            const v4f t = v[u];
            __builtin_nontemporal_store(t + 1.0f, o1v + i);
            __builtin_nontemporal_store(t - 1.0f, o2v + i);
            __builtin_nontemporal_store(t * 2.0f, o3v + i);
            __builtin_nontemporal_store(t * 0.5f, o4v + i);  // /2 == *0.5 exactly
            acc += t.x + t.y + t.z + t.w;
        }
    }

    // ---- Leftover v4f elements (none for N = 2^25, kept for generality) ----
    const unsigned gtid    = blockIdx.x * BLOCK + threadIdx.x;
    const unsigned gstride = gridDim.x * BLOCK;
    for (unsigned i = n4_full + gtid; i < n4; i += gstride) {
        const v4f t = __builtin_nontemporal_load(xv + i);
        __builtin_nontemporal_store(t + 1.0f, o1v + i);
        __builtin_nontemporal_store(t - 1.0f, o2v + i);
        __builtin_nontemporal_store(t * 2.0f, o3v + i);
        __builtin_nontemporal_store(t * 0.5f, o4v + i);
        acc += t.x + t.y + t.z + t.w;
    }

    // ---- Scalar tail (n % 4, none here) ----
    for (unsigned i = n4 * 4 + gtid; i < n; i += gstride) {
        const float v = x[i];
        o1[i] = v + 1.0f;
        o2[i] = v - 1.0f;
        o3[i] = v * 2.0f;
        o4[i] = v * 0.5f;
        acc += v;
    }

    // ---- Deterministic wave32 reduction (CDNA5 is wave32: offsets 16..1) ----
    #pragma unroll
    for (int off = 16; off > 0; off >>= 1)
        acc += __shfl_down(acc, off, 32);

    __shared__ float smem[WAVES_PER_BLOCK];
    const int lane = threadIdx.x & 31;
    const int wave = threadIdx.x >> 5;
    if (lane == 0) smem[wave] = acc;
    __syncthreads();

    if (wave == 0) {
        float s = (lane < WAVES_PER_BLOCK) ? smem[lane] : 0.0f;
        #pragma unroll
        for (int off = WAVES_PER_BLOCK / 2; off > 0; off >>= 1)
            s += __shfl_down(s, off, 32);
        if (lane == 0) partials[blockIdx.x] = s;  // fixed slot per block: deterministic
    }
}

__global__ __launch_bounds__(BLOCK) void reduce_partials_kernel(
    const float* __restrict__ partials, int nparts, float* __restrict__ out_sum)
{
    __shared__ float smem[BLOCK];
    float s = 0.0f;
    // Fixed per-thread stride order -> deterministic accumulation.
    for (int i = threadIdx.x; i < nparts; i += BLOCK)
        s += partials[i];
    smem[threadIdx.x] = s;
    __syncthreads();
    #pragma unroll
    for (int off = BLOCK / 2; off > 0; off >>= 1) {
        if ((int)threadIdx.x < off) smem[threadIdx.x] += smem[threadIdx.x + off];
        __syncthreads();
    }
    if (threadIdx.x == 0) *out_sum = smem[0];
}

extern "C" void kernel_launch(void* const* d_in, const int* in_sizes, int n_in,
                              void* d_out, int out_size, void* d_ws, size_t ws_size,
                              hipStream_t stream)
{
    const float* x = (const float*)d_in[0];
    const unsigned n  = (unsigned)in_sizes[0];      // 4096*8192 = 33,554,432
    const unsigned n4 = n / 4;                      // 8,388,608 v4f elements

    float* out = (float*)d_out;                     // [x1 | x2 | x3 | x4 | sum]
    float* o1 = out;
    float* o2 = out + (size_t)n;
    float* o3 = out + (size_t)2 * n;
    float* o4 = out + (size_t)3 * n;
    float* sum_out = out + (size_t)4 * n;

    float* partials = (float*)d_ws;

    // One tile per block at the exact size: 8,388,608 / 2048 = 4096 blocks.
    long long grid = ((long long)n4 + CHUNK - 1) / CHUNK;
    if (grid < 1) grid = 1;
    if (grid > 4096) grid = 4096;                   // grid-stride covers the rest
    long long ws_cap = (long long)(ws_size / sizeof(float));
    if (ws_cap >= 1 && grid > ws_cap) grid = ws_cap;

    fused_stream_kernel<<<(int)grid, BLOCK, 0, stream>>>(
        x, o1, o2, o3, o4, partials, n4, n);
    reduce_partials_kernel<<<1, BLOCK, 0, stream>>>(partials, (int)grid, sum_out);
}